// GenericLoss_15831249453099
// MI455X (gfx1250) — compile-verified
//
#include <hip/hip_runtime.h>

// Explicit ext-vector types (HIP's float4 struct is not valid for
// __builtin_nontemporal_load; clang ext_vector types are).
typedef float  v2f  __attribute__((ext_vector_type(2)));
typedef float  v4f  __attribute__((ext_vector_type(4)));
typedef float  v8f  __attribute__((ext_vector_type(8)));
typedef int    v4i  __attribute__((ext_vector_type(4)));

// log_sigmoid(-1) and log_sigmoid(0) = -ln2 (f32-rounded)
#define LSIG_M1 (-1.3132616875182228f)
#define MLN2    (-0.6931471805599453f)

// ---------------------------------------------------------------------------
// WMMA-based reduction of a 16x16 f32 matrix held in LDS (256 floats) down to
// its 16 column sums, using V_WMMA_F32_16X16X4_F32 with A = ones(16x4).
//
// B operand layout for 16x16x4 f32 (4x16 matrix, 2 VGPRs):
//   VGPR0: lanes 0-15 -> B[0][n], lanes 16-31 -> B[2][n]
//   VGPR1: lanes 0-15 -> B[1][n], lanes 16-31 -> B[3][n]
// Feeding row slab {4j..4j+3} of V per step and chaining D as accumulator
// gives D[m][n] = sum_{r=0..15} V[r][n] (column sums, replicated over rows m).
// Exact f32 adds; fixed order -> deterministic. EXEC must be all 1s: every
// wave in the block executes this uniformly.
// ---------------------------------------------------------------------------
__device__ __forceinline__ v8f block256_colsums_wmma(const float* __restrict__ s,
                                                     unsigned lane) {
  const unsigned col  = lane & 15u;
  const unsigned half = (lane >> 4) & 1u;
  v2f a;
  a[0] = 1.0f;
  a[1] = 1.0f;
  v8f d = {};
#pragma unroll
  for (unsigned j = 0; j < 4; ++j) {
    const unsigned r0 = 4u * j + 2u * half;   // rows {r0, r0+1} for this half
    v2f b;
    b[0] = s[r0 * 16u + col];
    b[1] = s[(r0 + 1u) * 16u + col];
    // 8 args: (neg_a, A, neg_b, B, c_mod, C, reuse_a, reuse_b)
    d = __builtin_amdgcn_wmma_f32_16x16x4_f32(false, a, false, b,
                                              (short)0, d, false, false);
  }
  return d;
}

// ---------------------------------------------------------------------------
// Kernel 1: streaming elementwise + per-block reduction.
//   contrib = mask * ( attr!=0 ? (score - 1.3132617) : -ln2 )
// Non-temporal b128 loads: 201 MB streamed once, > 192 MB L2.
// ---------------------------------------------------------------------------
__global__ __launch_bounds__(256)
void loss_partial_kernel(const int* __restrict__ attr,
                         const float* __restrict__ scores,
                         const float* __restrict__ mask,
                         float* __restrict__ partials,
                         int n4) {
  __shared__ float s_red[256];
  __shared__ float s_col[16];

  const int tid    = (int)threadIdx.x;
  const int gid    = (int)(blockIdx.x * blockDim.x + threadIdx.x);
  const int stride = (int)(gridDim.x * blockDim.x);

  const v4i* __restrict__ a4 = (const v4i*)attr;
  const v4f* __restrict__ s4 = (const v4f*)scores;
  const v4f* __restrict__ m4 = (const v4f*)mask;

  float acc = 0.0f;
  for (int i = gid; i < n4; i += stride) {
    const v4i a = __builtin_nontemporal_load(&a4[i]);
    const v4f s = __builtin_nontemporal_load(&s4[i]);
    const v4f m = __builtin_nontemporal_load(&m4[i]);
#pragma unroll
    for (int c = 0; c < 4; ++c) {
      const float term = (a[c] != 0) ? (s[c] + LSIG_M1) : MLN2;
      acc = __builtin_fmaf(m[c], term, acc);
    }
  }

  s_red[tid] = acc;
  __syncthreads();

  // All 8 waves execute the WMMA reduction uniformly (EXEC all ones).
  const v8f d = block256_colsums_wmma(s_red, (unsigned)(tid & 31));

  if (tid < 16) s_col[tid] = d[0];   // row 0 of D = column sums
  __syncthreads();

  if (tid == 0) {
    float t = 0.0f;
#pragma unroll
    for (int c = 0; c < 16; ++c) t += s_col[c];
    partials[blockIdx.x] = t;
  }
}

// ---------------------------------------------------------------------------
// Kernel 2: reduce the block partials to the final scalar, deterministic.
// ---------------------------------------------------------------------------
__global__ __launch_bounds__(256)
void loss_final_kernel(const float* __restrict__ partials, int np,
                       float* __restrict__ out, float scale) {
  __shared__ float s_red[256];
  __shared__ float s_col[16];

  const int tid = (int)threadIdx.x;

  float acc = 0.0f;
  for (int i = tid; i < np; i += 256) acc += partials[i];

  s_red[tid] = acc;
  __syncthreads();

  const v8f d = block256_colsums_wmma(s_red, (unsigned)(tid & 31));

  if (tid < 16) s_col[tid] = d[0];
  __syncthreads();

  if (tid == 0) {
    float t = 0.0f;
#pragma unroll
    for (int c = 0; c < 16; ++c) t += s_col[c];
    out[0] = t * scale;   // scale = -1/(B*C)
  }
}

// ---------------------------------------------------------------------------
// Launch: inputs are {attributes:int32[B*C], scores:f32[B*C], loss_mask:f32[B*C]}.
// Output: single f32 scalar. d_ws holds the 1024 block partials.
// ---------------------------------------------------------------------------
extern "C" void kernel_launch(void* const* d_in, const int* in_sizes, int n_in,
                              void* d_out, int out_size, void* d_ws, size_t ws_size,
                              hipStream_t stream) {
  (void)n_in; (void)out_size;

  const int* attr     = (const int*)d_in[0];
  const float* scores = (const float*)d_in[1];
  const float* mask   = (const float*)d_in[2];
  float* out          = (float*)d_out;
  float* partials     = (float*)d_ws;

  const int n  = in_sizes[0];       // B*C = 16,777,216 (multiple of 4)
  const int n4 = n >> 2;

  int nblocks = 1024;
  const size_t need = (size_t)nblocks * sizeof(float);
  if (ws_size < need) nblocks = (int)(ws_size / sizeof(float));
  if (nblocks < 1) nblocks = 1;

  loss_partial_kernel<<<nblocks, 256, 0, stream>>>(attr, scores, mask,
                                                   partials, n4);

  const float scale = -1.0f / (float)n;
  loss_final_kernel<<<1, 256, 0, stream>>>(partials, nblocks, out, scale);
}